// IntegrateWithWeights1DModule_89790586290716
// MI455X (gfx1250) — compile-verified
//
#include <hip/hip_runtime.h>
#include <stdint.h>

#define THREADS 256
#define WPB (THREADS / 32)      // waves per block (wave32)
#define MAIN_BLOCKS 2048

// ------------------------------------------------------------------
// Zero the output (harness poisons d_out; atomics need a clean base).
__global__ void zero_out_kernel(float* __restrict__ out, int n) {
  int i = blockIdx.x * blockDim.x + threadIdx.x;
  if (i < n) out[i] = 0.0f;
}

// CDNA5 async copy: 16 bytes per lane, global -> LDS, tracked by ASYNCcnt.
__device__ __forceinline__ void async_copy_b128(unsigned lds_byte_addr, const void* gptr) {
  asm volatile("global_load_async_to_lds_b128 %0, %1, off"
               :
               : "v"(lds_byte_addr), "v"(gptr)
               : "memory");
}

// Low 32 bits of a flat shared pointer are the LDS byte address (ISA 10.2).
__device__ __forceinline__ unsigned lds_addr32(const void* p) {
  return (unsigned)(uintptr_t)p;
}

// DPP lane-shift (shfl_up within 16-lane rows, zero-fill), VALU not DS.
template <int CTRL>
__device__ __forceinline__ float dpp_shr_f32(float x) {
  return __int_as_float(
      __builtin_amdgcn_update_dpp(0, __float_as_int(x), CTRL, 0xF, 0xF, false));
}

// Wave32 inclusive prefix sum: 4 DPP adds + readlane(15) + predicated add.
__device__ __forceinline__ float wave32_inclusive_scan(float x, int lane) {
  x += dpp_shr_f32<0x111>(x);   // row_shr:1
  x += dpp_shr_f32<0x112>(x);   // row_shr:2
  x += dpp_shr_f32<0x114>(x);   // row_shr:4
  x += dpp_shr_f32<0x118>(x);   // row_shr:8
  const float row0 = __int_as_float(__builtin_amdgcn_readlane(__float_as_int(x), 15));
  if (lane >= 16) x += row0;
  return x;
}

// ------------------------------------------------------------------
// Each wave streams contiguous 128-sample windows through LDS (double
// buffered async copies), then does 4 wave32 segmented reductions; only
// run-end lanes issue a global f32 atomic.
__global__ __launch_bounds__(THREADS)
void seg_integrate_kernel(const int* __restrict__ seg_ids,
                          const float* __restrict__ values,
                          const float* __restrict__ weights,
                          float* __restrict__ out,
                          int n, int full_windows)
{
  __shared__ __align__(16) int   s_id[WPB * 2 * 128];
  __shared__ __align__(16) float s_v [WPB * 2 * 128];
  __shared__ __align__(16) float s_w [WPB * 2 * 128];

  const int tid    = threadIdx.x;
  const int lane   = tid & 31;
  const int wib    = tid >> 5;                       // wave index in block
  const int wgid   = blockIdx.x * WPB + wib;         // global wave index
  const int nwaves = gridDim.x * WPB;

  int*   li = s_id + wib * 256;                      // 2 buffers x 128
  float* lv = s_v  + wib * 256;
  float* lw = s_w  + wib * 256;

  // Stage window W into LDS buffer p: each lane moves 16B of each stream.
  auto issue = [&](int W, int p) {
    const size_t s   = (size_t)W * 128 + (size_t)lane * 4;
    const int    off = p * 128 + lane * 4;
    async_copy_b128(lds_addr32(li + off), (const void*)(seg_ids + s));
    async_copy_b128(lds_addr32(lv + off), (const void*)(values  + s));
    async_copy_b128(lds_addr32(lw + off), (const void*)(weights + s));
  };

  int W = wgid;
  int p = 0;
  if (W < full_windows) issue(W, 0);

  while (W < full_windows) {
    const int Wn = W + nwaves;
    if (Wn < full_windows) {
      issue(Wn, p ^ 1);                               // prefetch next window
      asm volatile("s_wait_asynccnt 3" ::: "memory"); // current 3 copies done
    } else {
      asm volatile("s_wait_asynccnt 0" ::: "memory");
    }

    // Each lane reads back exactly the 16B-per-stream it staged.
    const int off = p * 128 + lane * 4;
    const int4   i4 = *reinterpret_cast<const int4*>  (li + off);
    const float4 v4 = *reinterpret_cast<const float4*>(lv + off);
    const float4 w4 = *reinterpret_cast<const float4*>(lw + off);

    const int   ida[4] = {i4.x, i4.y, i4.z, i4.w};
    const float ca [4] = {v4.x * w4.x, v4.y * w4.y, v4.z * w4.z, v4.w * w4.w};

#pragma unroll
    for (int j = 0; j < 4; ++j) {
      // Sub-round j covers samples 4*lane+j: ids monotonic in lane order.
      const int   id = ida[j];
      const float c  = ca[j];

      // Previous-lane id: DPP row_shr:1, patch lane 16 via readlane(15).
      int idp = __builtin_amdgcn_update_dpp(0, id, 0x111, 0xF, 0xF, false);
      const int id15 = __builtin_amdgcn_readlane(id, 15);
      if (lane == 16) idp = id15;
      const bool head = (lane == 0) || (idp != id);
      const unsigned mask = (unsigned)__ballot(head);

      const float sc = wave32_inclusive_scan(c, lane);

      // Segment sum = scan - scan at (run start - 1).
      const unsigned below    = mask & ((2u << lane) - 1u);
      const int      runstart = 31 - __builtin_clz(below);
      const float    pprev    = __shfl(sc, (runstart > 0) ? (runstart - 1) : 0);
      const float    seg      = sc - ((runstart > 0) ? pprev : 0.0f);

      // Only the last lane of each run flushes.
      const bool flush = (lane == 31) || ((mask >> (lane + 1)) & 1u);
      if (flush) atomicAdd(&out[id], seg);
    }

    W = Wn;
    p ^= 1;
  }

  // Remainder (n % 128) — never taken for N = 16,777,216, kept for safety.
  if (wgid == 0) {
    for (int s = full_windows * 128 + lane; s < n; s += 32)
      atomicAdd(&out[seg_ids[s]], values[s] * weights[s]);
  }
}

// ------------------------------------------------------------------
extern "C" void kernel_launch(void* const* d_in, const int* in_sizes, int n_in,
                              void* d_out, int out_size, void* d_ws, size_t ws_size,
                              hipStream_t stream) {
  const int*   seg_ids = (const int*)  d_in[0];
  const float* values  = (const float*)d_in[1];
  const float* weights = (const float*)d_in[2];
  float* out = (float*)d_out;

  const int n            = in_sizes[0];
  const int full_windows = n / 128;

  zero_out_kernel<<<(out_size + THREADS - 1) / THREADS, THREADS, 0, stream>>>(out, out_size);
  seg_integrate_kernel<<<MAIN_BLOCKS, THREADS, 0, stream>>>(
      seg_ids, values, weights, out, n, full_windows);
}